// FullAttention_71657234366635
// MI455X (gfx1250) — compile-verified
//
#include <hip/hip_runtime.h>

typedef float v2f __attribute__((ext_vector_type(2)));
typedef float v8f __attribute__((ext_vector_type(8)));

#define L_SEQ    2048
#define DH       32
#define NHEADS   8
#define SBLK     64
#define NEGV     (-1.0e8f)
#define SM_SCALE 0.17677669529663689f   /* 1/sqrt(32) */

// One workgroup = (n, h, 128-query block); 8 waves of 32, each wave owns 16 query rows.
__global__ __launch_bounds__(256)
void fullattn_f32wmma_kernel(const float* __restrict__ q,
                             const float* __restrict__ k,
                             const float* __restrict__ v,
                             const int*   __restrict__ mask0,
                             const int*   __restrict__ mask1,
                             float*       __restrict__ out)
{
    __shared__ __align__(16) float LK[DH * SBLK];        // K tile   [d][s]
    __shared__ __align__(16) float LV[DH * SBLK];        // V tile   [d][s]
    __shared__ __align__(16) float LP[8][16 * SBLK];     // per-wave P scratch [m][s]
    __shared__ int LM1[SBLK];

    const int tid    = threadIdx.x;
    const int lane   = tid & 31;
    const int wave   = tid >> 5;
    const int half   = lane >> 4;      // 0: lanes 0-15, 1: lanes 16-31
    const int lane16 = lane & 15;

    const int bx   = blockIdx.x;
    const int mblk = bx & 15;          // 16 blocks of 128 queries
    const int h    = (bx >> 4) & 7;
    const int n    = bx >> 7;

    const int l0 = mblk * 128 + wave * 16;   // this wave's first query row
    const size_t headoff = ((size_t)n * NHEADS + h) * (size_t)DH * L_SEQ;
    const float* qh = q + headoff;
    const float* kh = k + headoff;
    const float* vh = v + headoff;

    // ---- Q tile in WMMA A-layout registers (M=16, per K-chunk of 4):
    //      a[2c+j] = Q^T[M=lane16][K=4c + 2*half + j] = q[d][l0+lane16]
    float qa[16];
#pragma unroll
    for (int c = 0; c < 8; ++c) {
        const int d0 = 4 * c + 2 * half;
        qa[2*c+0] = qh[(size_t)(d0 + 0) * L_SEQ + l0 + lane16];
        qa[2*c+1] = qh[(size_t)(d0 + 1) * L_SEQ + l0 + lane16];
    }

    const int m0r = mask0[(size_t)n * L_SEQ + l0 + lane16];  // this lane's query-row mask

    v8f oc0 = {};          // O accumulator, d = 0..15  (C-layout)
    v8f oc1 = {};          // O accumulator, d = 16..31
    float m_run = -INFINITY;
    float l_run = 0.0f;

    float* myP = &LP[wave][0];

    for (int s0 = 0; s0 < L_SEQ; s0 += SBLK) {
        __syncthreads();   // protect LK/LV against reuse across iterations
        {   // cooperative K/V stage: 32x64 floats each, 8 floats (2x float4) per thread
            const int d = tid >> 3;
            const int s = (tid & 7) * 8;
            const float4* kg = (const float4*)(kh + (size_t)d * L_SEQ + s0 + s);
            const float4* vg = (const float4*)(vh + (size_t)d * L_SEQ + s0 + s);
            float4* lk4 = (float4*)&LK[d * SBLK + s];
            float4* lv4 = (float4*)&LV[d * SBLK + s];
            lk4[0] = kg[0]; lk4[1] = kg[1];
            lv4[0] = vg[0]; lv4[1] = vg[1];
            if (tid < SBLK) LM1[tid] = mask1[(size_t)n * L_SEQ + s0 + tid];
        }
        __syncthreads();

        // ---- S = Q^T K : 16 x 64 as four 16x16 WMMA tiles, K-dim = 32 (8 chunks of 4)
#pragma unroll
        for (int nt = 0; nt < 4; ++nt) {
            v8f acc = {};
#pragma unroll
            for (int c = 0; c < 8; ++c) {
                const int d0 = 4 * c + 2 * half;
                v2f a; a.x = qa[2*c+0]; a.y = qa[2*c+1];
                v2f b;
                b.x = LK[(d0 + 0) * SBLK + nt * 16 + lane16];   // B[K=d0  ][N=lane16]
                b.y = LK[(d0 + 1) * SBLK + nt * 16 + lane16];   // B[K=d0+1][N=lane16]
                acc = __builtin_amdgcn_wmma_f32_16x16x4_f32(
                          false, a, false, b, (short)0, acc, false, false);
            }
            // spill C-layout tile: VGPR r -> row r + 8*half, col nt*16 + lane16
#pragma unroll
            for (int r = 0; r < 8; ++r)
                myP[(r + 8 * half) * SBLK + nt * 16 + lane16] = acc[r];
        }
        __syncthreads();   // P visible across lanes

        // ---- online softmax over this 64-key block (lane: row=lane16, cols half*32..+31)
        const int row   = lane16;
        const int cbase = half * 32;
        float mblk_loc = -INFINITY;
#pragma unroll
        for (int j = 0; j < 32; ++j) {
            const float val   = myP[row * SBLK + cbase + j];
            const int   valid = m0r & LM1[cbase + j];
            const float sv    = (valid ? val : NEGV) * SM_SCALE;
            myP[row * SBLK + cbase + j] = sv;
            mblk_loc = fmaxf(mblk_loc, sv);
        }
        const float mblk_row = fmaxf(mblk_loc, __shfl_xor(mblk_loc, 16, 32));
        const float m_new    = fmaxf(m_run, mblk_row);
        const float alpha    = __expf(m_run - m_new);

        float lblk_loc = 0.0f;
#pragma unroll
        for (int j = 0; j < 32; ++j) {
            const float sv = myP[row * SBLK + cbase + j];
            const float e  = __expf(sv - m_new);
            myP[row * SBLK + cbase + j] = e;
            lblk_loc += e;
        }
        l_run = l_run * alpha + (lblk_loc + __shfl_xor(lblk_loc, 16, 32));
        m_run = m_new;

        // rescale O accumulators: row of VGPR r is (r + 8*half); alpha lives at lane==row
#pragma unroll
        for (int r = 0; r < 8; ++r) {
            const float ar = __shfl(alpha, r + 8 * half, 32);
            oc0[r] *= ar;
            oc1[r] *= ar;
        }

        __syncthreads();   // exp(P) stores visible before cross-lane A-layout reads

        // ---- O += P * V^T : K-dim = 64 (16 chunks of 4), two 16-wide d tiles
#pragma unroll
        for (int c = 0; c < 16; ++c) {
            const int sA = 4 * c + 2 * half;
            v2f a;                                        // A[M=lane16][K=sA+j]
            a.x = myP[lane16 * SBLK + sA + 0];
            a.y = myP[lane16 * SBLK + sA + 1];
            v2f b0, b1;                                   // B[K=sA+j][N=d] = v[d][s]
            b0.x = LV[(lane16 +  0) * SBLK + sA + 0];
            b0.y = LV[(lane16 +  0) * SBLK + sA + 1];
            b1.x = LV[(lane16 + 16) * SBLK + sA + 0];
            b1.y = LV[(lane16 + 16) * SBLK + sA + 1];
            oc0 = __builtin_amdgcn_wmma_f32_16x16x4_f32(
                      false, a, false, b0, (short)0, oc0, false, false);
            oc1 = __builtin_amdgcn_wmma_f32_16x16x4_f32(
                      false, a, false, b1, (short)0, oc1, false, false);
        }
    }

    // ---- normalize rows and store: out[n][h*32 + d][l0 + M], d-major stride L
    float* outp = out + headoff;
#pragma unroll
    for (int r = 0; r < 8; ++r) {
        const float inv = 1.0f / __shfl(l_run, r + 8 * half, 32);
        const int   M   = r + 8 * half;
        outp[(size_t)(lane16 +  0) * L_SEQ + l0 + M] = oc0[r] * inv;
        outp[(size_t)(lane16 + 16) * L_SEQ + l0 + M] = oc1[r] * inv;
    }
}

extern "C" void kernel_launch(void* const* d_in, const int* in_sizes, int n_in,
                              void* d_out, int out_size, void* d_ws, size_t ws_size,
                              hipStream_t stream) {
    const float* q  = (const float*)d_in[0];
    const float* k  = (const float*)d_in[1];
    const float* v  = (const float*)d_in[2];
    const int*   m0 = (const int*)d_in[3];
    const int*   m1 = (const int*)d_in[4];
    float* out = (float*)d_out;

    const int N = in_sizes[0] / (256 * L_SEQ);   // q is (N, 256, L)
    dim3 grid(N * NHEADS * (L_SEQ / 128));       // 512 workgroups for N=4
    dim3 block(256);
    fullattn_f32wmma_kernel<<<grid, block, 0, stream>>>(q, k, v, m0, m1, out);
}